// SConv2d_67250597921090
// MI455X (gfx1250) — compile-verified
//
#include <hip/hip_runtime.h>
#include <stdint.h>

// ---------------------------------------------------------------------------
// SConv2d majority-gate "conv": per output pixel, 243 products (C,kh,kw order,
// pad = -1) reduced by a 5-level 3-ary majority tree.
// s-domain identity: level-0 s = (x*w+1)/2 = x*(w/2) + 0.5; every gate
// r = ab+ac+bc-2abc stays in s-domain (the reference's 2r-1 / (y+1)/2 maps
// cancel between levels); final output y = 2r-1.
// Weights are pre-scaled by 0.5 at staging time, so level-0 is one FMA/elem.
// Gate costs 1 mul + 1 add + 2 fma.  x halo tile staged via the CDNA5
// async global->LDS path (ASYNCcnt / s_wait_asynccnt).
// ---------------------------------------------------------------------------

#define GLOBAL_AS __attribute__((address_space(1)))
#define LDS_AS    __attribute__((address_space(3)))

#define C_IN  27
#define NOC   27
#define HALO  18
#define XT    (C_IN * HALO * HALO)   // 8748 floats = 34992 B in LDS

__device__ __forceinline__ void async_copy_f32(const float* g, float* l) {
#if __has_builtin(__builtin_amdgcn_global_load_async_to_lds_b32)
  __builtin_amdgcn_global_load_async_to_lds_b32(
      (GLOBAL_AS int*)g, (LDS_AS int*)l, /*imm offset*/0, /*cpol*/0);
#else
  unsigned lds_off = (unsigned)(uintptr_t)(LDS_AS float*)l;
  unsigned long long ga = (unsigned long long)(uintptr_t)g;
  asm volatile("global_load_async_to_lds_b32 %0, %1, off"
               :: "v"(lds_off), "v"(ga) : "memory");
#endif
}

__device__ __forceinline__ void wait_async_zero() {
#if __has_builtin(__builtin_amdgcn_s_wait_asynccnt)
  __builtin_amdgcn_s_wait_asynccnt(0);
#else
  asm volatile("s_wait_asynccnt 0" ::: "memory");
#endif
}

// maj3 in s-domain: ab + ac + bc - 2abc = t + c*(u - 2t), t=ab, u=a+b
__device__ __forceinline__ float maj3(float a, float b, float c) {
  float t = a * b;
  float u = a + b;
  return fmaf(c, fmaf(-2.0f, t, u), t);
}

__global__ __launch_bounds__(256)
void sconv2d_maj3_kernel(const float* __restrict__ x,
                         const float* __restrict__ w,
                         float* __restrict__ out) {
  __shared__ float xs[XT];          // [c][18][18] halo tile, pad = -1
  __shared__ float ws[C_IN * 9];    // 0.5 * weights for this oc, (c,kh,kw)

  const int b   = blockIdx.x;
  const int t   = b & 3;                 // spatial 16x16 tile: 0..3
  const int oc  = (b >> 2) % NOC;
  const int n   = b / (4 * NOC);
  const int r0  = (t >> 1) << 4;
  const int c0  = (t & 1) << 4;
  const int tid = threadIdx.x;

  // Phase 1: fill halo tile with the unfold pad value (-1).
  for (int i = tid; i < XT; i += 256) xs[i] = -1.0f;
  __syncthreads();

  // Phase 2a: async-stage in-bounds interior of the 27 x 18 x 18 halo tile.
  const float* xn = x + n * (C_IN * 32 * 32);
  for (int i = tid; i < XT; i += 256) {
    int c  = i / (HALO * HALO);
    int rm = i % (HALO * HALO);
    int li = rm / HALO;
    int lc = rm - li * HALO;
    int rg = r0 - 1 + li;
    int cg = c0 - 1 + lc;
    if ((unsigned)rg < 32u && (unsigned)cg < 32u)
      async_copy_f32(xn + (c * 32 + rg) * 32 + cg, &xs[i]);
  }
  // Phase 2b: stage half-weights (regular load + ds store; ordered by barrier).
  if (tid < C_IN * 9)
    ws[tid] = 0.5f * w[oc * (C_IN * 9) + tid];

  wait_async_zero();   // my wave's async LDS writes have landed
  __syncthreads();     // everyone's writes (async + ds) visible

  // Phase 3: 3^5 majority tree per pixel, reading the shared halo tile.
  const int tx = tid & 15;
  const int ty = tid >> 4;
  const float* xw = &xs[ty * HALO + tx];

  float m4[3];
#pragma unroll
  for (int i4 = 0; i4 < 3; ++i4) {
    float m3[3];
#pragma unroll
    for (int i3 = 0; i3 < 3; ++i3) {
      float m2[3];
#pragma unroll
      for (int i2 = 0; i2 < 3; ++i2) {
        const int c = (i4 * 3 + i3) * 3 + i2;
        const float* xc = xw + c * (HALO * HALO);
        const float* wc = &ws[c * 9];
        float m1[3];
#pragma unroll
        for (int kh = 0; kh < 3; ++kh) {
          // s = x*(w/2) + 0.5  -> one FMA per element
          float s0 = fmaf(xc[kh * HALO + 0], wc[kh * 3 + 0], 0.5f);
          float s1 = fmaf(xc[kh * HALO + 1], wc[kh * 3 + 1], 0.5f);
          float s2 = fmaf(xc[kh * HALO + 2], wc[kh * 3 + 2], 0.5f);
          m1[kh] = maj3(s0, s1, s2);
        }
        m2[i2] = maj3(m1[0], m1[1], m1[2]);
      }
      m3[i3] = maj3(m2[0], m2[1], m2[2]);
    }
    m4[i4] = maj3(m3[0], m3[1], m3[2]);
  }
  float r = maj3(m4[0], m4[1], m4[2]);

  out[((n * NOC + oc) * 32 + (r0 + ty)) * 32 + (c0 + tx)] = 2.0f * r - 1.0f;
}

extern "C" void kernel_launch(void* const* d_in, const int* in_sizes, int n_in,
                              void* d_out, int out_size, void* d_ws, size_t ws_size,
                              hipStream_t stream) {
  (void)n_in; (void)out_size; (void)d_ws; (void)ws_size;
  const float* x = (const float*)d_in[0];   // (N, 27, 32, 32) f32
  const float* w = (const float*)d_in[1];   // (27, 27, 3, 3)  f32
  float* out = (float*)d_out;               // (N, 27, 32, 32) f32

  const int N = in_sizes[0] / (C_IN * 32 * 32);   // = 8 for this harness
  dim3 grid(N * NOC * 4), block(256);
  sconv2d_maj3_kernel<<<grid, block, 0, stream>>>(x, w, out);
}